// CSDBF_35064113005082
// MI455X (gfx1250) — compile-verified
//
#include <hip/hip_runtime.h>
#include <hip/hip_bf16.h>
#include <math.h>

// ---------------------------------------------------------------------------
// GAT forward for MI455X (gfx1250, wave32).
// att@Wh runs on V_WMMA_F32_16X16X4_F32 (fp32 matrix pipe).
// Adjacency (256MB int32) is read exactly once and packed to bits (8MB, L2).
// Wh is stored pair-interleaved so WMMA B operands load with zero reg moves.
// Inner loop is software-pipelined one iteration ahead; softmax 1/Z is
// applied once per output element instead of per attention weight.
// ---------------------------------------------------------------------------

typedef float v2f __attribute__((ext_vector_type(2)));
typedef float v8f __attribute__((ext_vector_type(8)));

#define GN1 3000
#define GN2 2500
#define GN3 2500
#define NTOT 8000
#define NEGBIG -9000000000000000.0f
#define LALPHA 0.2f
#define MWORDS 256   /* padded words per mask row (1KB stride) */
#define MUSED  250   /* 8000/32 words actually used            */

// ---------------------------------------------------------------------------
// Pass 0: pack the 9 adjacency blocks into a bit matrix. One thread per
// 32-bit output word (32 consecutive j for one row); 256MB read once.
// ---------------------------------------------------------------------------
__global__ __launch_bounds__(256) void gat_pack_mask(
    const int* __restrict__ A1,  const int* __restrict__ A2,  const int* __restrict__ A3,
    const int* __restrict__ A12, const int* __restrict__ A13, const int* __restrict__ A23,
    const int* __restrict__ A21, const int* __restrict__ A31, const int* __restrict__ A32,
    unsigned int* __restrict__ mbits)
{
  int id = blockIdx.x * 256 + threadIdx.x;
  if (id >= NTOT * MUSED) return;
  int row  = id / MUSED;
  int word = id - row * MUSED;
  int j0 = word * 32;
  unsigned int bits = 0u;
  #pragma unroll 4
  for (int b = 0; b < 32; ++b) {
    int j = j0 + b;               // j <= 7999, no bound check needed
    int v;
    if (row < GN1) {
      if (j < GN1)            v = A1 [row * GN1 + j];
      else if (j < GN1 + GN2) v = A12[row * GN2 + (j - GN1)];
      else                    v = A13[row * GN3 + (j - GN1 - GN2)];
    } else if (row < GN1 + GN2) {
      int r = row - GN1;
      if (j < GN1)            v = A21[r * GN1 + j];
      else if (j < GN1 + GN2) v = A2 [r * GN2 + (j - GN1)];
      else                    v = A23[r * GN3 + (j - GN1 - GN2)];
    } else {
      int r = row - GN1 - GN2;
      if (j < GN1)            v = A31[r * GN1 + j];
      else if (j < GN1 + GN2) v = A32[r * GN2 + (j - GN1)];
      else                    v = A3 [r * GN3 + (j - GN1 - GN2)];
    }
    bits |= (v != 0 ? 1u : 0u) << b;
  }
  mbits[row * MWORDS + word] = bits;
}

// ---------------------------------------------------------------------------
// Pass 1: Wh[h][n][o] = sum_f h[n][f]*Ws[h][f][o]; s = Wh.a1, t = Wh.a2.
// Wh is written PAIR-INTERLEAVED:  WhI[h][n>>1][o*2 + (n&1)]
// so that pass 3's B operands (v2f over two consecutive j) are contiguous.
// ---------------------------------------------------------------------------
__global__ __launch_bounds__(128) void gat_wh(
    const float* __restrict__ h, const float* __restrict__ Ws, const float* __restrict__ ap,
    float* __restrict__ WhI, float* __restrict__ s1, float* __restrict__ s2)
{
  __shared__ float hrow[128];
  __shared__ float whs[128];
  int n = blockIdx.x, tid = threadIdx.x;
  hrow[tid] = h[n * 128 + tid];
  __syncthreads();
  int head = tid >> 6, o = tid & 63;
  const float* W = Ws + head * (128 * 64) + o;
  float acc = 0.f;
  #pragma unroll 8
  for (int f = 0; f < 128; ++f) acc = fmaf(hrow[f], W[f * 64], acc);
  WhI[head * ((NTOT / 2) * 128) + (n >> 1) * 128 + o * 2 + (n & 1)] = acc;
  whs[tid] = acc;
  __syncthreads();
  if (tid < 4) {                       // 2 heads x {a1,a2}
    int hd = tid >> 1, sel = tid & 1;
    float acc2 = 0.f;
    for (int q = 0; q < 64; ++q)
      acc2 = fmaf(whs[hd * 64 + q], ap[hd * 128 + sel * 64 + q], acc2);
    (sel ? s2 : s1)[hd * NTOT + n] = acc2;
  }
}

// ---------------------------------------------------------------------------
// Pass 2: per (head,row) softmax stats. Two sweeps (max, then sum of exp):
// 1 exp per element; both sweeps hit L1/L2 (t row = 32KB, mask row = 1KB).
// One wave32 per row. Uses the reference's finite -9e15 sentinel so
// fully-masked rows reduce to uniform exactly like JAX.
// ---------------------------------------------------------------------------
__global__ __launch_bounds__(256) void gat_rowstats(
    const float* __restrict__ s1, const float* __restrict__ s2,
    const unsigned int* __restrict__ mbits,
    float* __restrict__ rowM, float* __restrict__ rowIZ)
{
  int w    = blockIdx.x * 8 + (threadIdx.x >> 5);
  int lane = threadIdx.x & 31;
  int head = w / NTOT;
  int row  = w - head * NTOT;
  float si = s1[head * NTOT + row];
  const float* tt = s2 + head * NTOT;
  const unsigned int* mrow = mbits + row * MWORDS;

  // sweep 1: row max of masked leaky-relu scores
  float m = NEGBIG;
  for (int k = 0; k < MUSED; ++k) {
    unsigned int wbits = mrow[k];          // broadcast (all lanes same word)
    float tj = tt[k * 32 + lane];
    float x = si + tj;
    float e = x > 0.f ? x : LALPHA * x;
    float val = ((wbits >> lane) & 1u) ? e : NEGBIG;
    m = fmaxf(m, val);
  }
  #pragma unroll
  for (int off = 16; off >= 1; off >>= 1) m = fmaxf(m, __shfl_xor(m, off));

  // sweep 2: Z = sum exp(val - m)  (masked-out terms underflow to 0)
  float Z = 0.f;
  for (int k = 0; k < MUSED; ++k) {
    unsigned int wbits = mrow[k];
    float tj = tt[k * 32 + lane];
    float x = si + tj;
    float e = x > 0.f ? x : LALPHA * x;
    float val = ((wbits >> lane) & 1u) ? e : NEGBIG;
    Z += __expf(val - m);
  }
  #pragma unroll
  for (int off = 16; off >= 1; off >>= 1) Z += __shfl_xor(Z, off);

  if (lane == 0) {
    rowM [head * NTOT + row] = m;
    rowIZ[head * NTOT + row] = 1.f / Z;    // Z >= 1 always
  }
}

// ---------------------------------------------------------------------------
// Pass 3: mid = ELU( softmax(e) @ Wh ) via V_WMMA_F32_16X16X4_F32.
// Block = 256 thr (8 waves) owns a (16-row, head) 16x64 output tile.
// A tile (16x4 fp32): lane holds M=lane%16, K = v + 2*(lane/16)  [ISA 7.12.2]
// B tiles: tile t covers output columns 4n+t; with pair-interleaved WhI one
// lane's 8 B floats are contiguous -> two b128 loads ARE the four v2f pairs.
// Software-pipelined: iteration it consumes data prefetched at it-1, so the
// s_wait before the WMMAs covers a full iteration of latency. The one-ahead
// prefetch at it=249 reads workspace padding (always in-bounds).
// A holds UNNORMALIZED exp(val-m); 1/Z applied once in the reduction.
// ---------------------------------------------------------------------------
__global__ __launch_bounds__(256) void gat_attn_wmma(
    const float* __restrict__ WhI, const float* __restrict__ s1, const float* __restrict__ s2,
    const float* __restrict__ rowM, const float* __restrict__ rowIZ,
    const unsigned int* __restrict__ mbits,
    float* __restrict__ mid)
{
  __shared__ float red[8][1024];           // 8 wave partials x (16x64) = 32KB
  int ib   = blockIdx.x >> 1;
  int head = blockIdx.x & 1;
  int tid  = threadIdx.x;
  int wid  = tid >> 5, lane = tid & 31;
  int lm = lane & 15, hi = lane >> 4;
  int row = ib * 16 + lm;                  // this lane's A-matrix row
  float si = s1  [head * NTOT + row];
  float mi = rowM[head * NTOT + row];

  // strength-reduced pointers; wave 'wid' covers k-chunks wid, wid+8, ...
  int bitpos = 4 * wid + 2 * hi;                       // 0..30, even
  const unsigned int* mp = mbits + row * MWORDS;       // word idx == iteration
  const float2* tp = (const float2*)(s2 + head * NTOT + bitpos);
  const float4* bp = (const float4*)(WhI + head * ((NTOT / 2) * 128)
                                         + (2 * wid + hi) * 128 + 8 * lm);

  // pipeline prologue
  float4 qa = bp[0], qb = bp[1];
  float2 tj = tp[0];
  unsigned int wb = mp[0];

  v8f c0 = {}, c1 = {}, c2 = {}, c3 = {};
  #pragma unroll 2
  for (int it = 0; it < 250; ++it) {       // 250 iters x 32 j = 8000
    // prefetch next iteration (it=249 touches padded workspace: safe)
    float4 qa_n = bp[512 * (it + 1)];
    float4 qb_n = bp[512 * (it + 1) + 1];
    float2 tj_n = tp[16 * (it + 1)];
    unsigned int wb_n = mp[it + 1];

    v2f A, B0, B1, B2, B3;
    B0[0] = qa.x; B0[1] = qa.y;
    B1[0] = qa.z; B1[1] = qa.w;
    B2[0] = qb.x; B2[1] = qb.y;
    B3[0] = qb.z; B3[1] = qb.w;
    float x0 = si + tj.x;
    float e0 = x0 > 0.f ? x0 : LALPHA * x0;
    float v0 = ((wb >> bitpos) & 1u) ? e0 : NEGBIG;
    A[0] = __expf(v0 - mi);                // unnormalized attention weight
    float x1 = si + tj.y;
    float e1 = x1 > 0.f ? x1 : LALPHA * x1;
    float v1 = ((wb >> (bitpos + 1)) & 1u) ? e1 : NEGBIG;
    A[1] = __expf(v1 - mi);
    c0 = __builtin_amdgcn_wmma_f32_16x16x4_f32(false, A, false, B0, (short)0, c0, false, false);
    c1 = __builtin_amdgcn_wmma_f32_16x16x4_f32(false, A, false, B1, (short)0, c1, false, false);
    c2 = __builtin_amdgcn_wmma_f32_16x16x4_f32(false, A, false, B2, (short)0, c2, false, false);
    c3 = __builtin_amdgcn_wmma_f32_16x16x4_f32(false, A, false, B3, (short)0, c3, false, false);

    qa = qa_n; qb = qb_n; tj = tj_n; wb = wb_n;
  }
  // C layout: VGPR r of lane -> (row_local = r + 8*hi, N = lane%16); tile t -> col 4N+t
  #pragma unroll
  for (int r = 0; r < 8; ++r) {
    int rl = r + 8 * hi;
    float4 f4 = { c0[r], c1[r], c2[r], c3[r] };
    *(float4*)(&red[wid][rl * 64 + 4 * lm]) = f4;
  }
  __syncthreads();
  const float* izr = rowIZ + head * NTOT + ib * 16;
  float* midh = mid + head * (NTOT * 64) + ib * (16 * 64);
  #pragma unroll
  for (int p = tid; p < 1024; p += 256) {
    float v = 0.f;
    #pragma unroll
    for (int wp = 0; wp < 8; ++wp) v += red[wp][p];
    v *= izr[p >> 6];                      // softmax 1/Z, once per output
    midh[p] = v > 0.f ? v : expm1f(v);     // ELU fused here; coalesced store
  }
}

// ---------------------------------------------------------------------------
// Pass 4: row-wise log_softmax over the 128 (=2 heads x 64) features, with
// the x3|x4|x5 output block permutation. One wave per row, float4 per lane.
// ---------------------------------------------------------------------------
__global__ __launch_bounds__(256) void gat_logsoftmax(
    const float* __restrict__ mid, float* __restrict__ out)
{
  int row  = blockIdx.x * 8 + (threadIdx.x >> 5);
  int lane = threadIdx.x & 31;
  int c = lane * 4;                        // feature index 0..127
  int head = c >> 6, col = c & 63;
  const float4 q = *(const float4*)(mid + head * (NTOT * 64) + row * 64 + col);
  float mx = fmaxf(fmaxf(q.x, q.y), fmaxf(q.z, q.w));
  #pragma unroll
  for (int off = 16; off >= 1; off >>= 1) mx = fmaxf(mx, __shfl_xor(mx, off));
  float sum = __expf(q.x - mx) + __expf(q.y - mx) + __expf(q.z - mx) + __expf(q.w - mx);
  #pragma unroll
  for (int off = 16; off >= 1; off >>= 1) sum += __shfl_xor(sum, off);
  float L = mx + __logf(sum);
  size_t base;
  if (row < GN1)            base = (size_t)GN3 * 128 + (size_t)row * 128;                 // x4
  else if (row < GN1 + GN2) base = (size_t)(GN3 + GN1) * 128 + (size_t)(row - GN1) * 128; // x5
  else                      base = (size_t)(row - GN1 - GN2) * 128;                       // x3
  float4 o4 = { q.x - L, q.y - L, q.z - L, q.w - L };
  *(float4*)(out + base + c) = o4;
}

// ---------------------------------------------------------------------------
extern "C" void kernel_launch(void* const* d_in, const int* in_sizes, int n_in,
                              void* d_out, int out_size, void* d_ws, size_t ws_size,
                              hipStream_t stream)
{
  const float* h   = (const float*)d_in[0];
  const int*   A1  = (const int*)d_in[1];
  const int*   A2  = (const int*)d_in[2];
  const int*   A3  = (const int*)d_in[3];
  const int*   A12 = (const int*)d_in[4];
  const int*   A13 = (const int*)d_in[5];
  const int*   A23 = (const int*)d_in[6];
  const int*   A21 = (const int*)d_in[7];
  const int*   A31 = (const int*)d_in[8];
  const int*   A32 = (const int*)d_in[9];
  const float* Ws  = (const float*)d_in[10];
  const float* ap  = (const float*)d_in[11];
  float* out = (float*)d_out;

  // workspace layout (bytes): WhI 4MB | mid 4MB | maskbits 8MB | 4x64KB vectors
  char* ws = (char*)d_ws;
  float*        WhI   = (float*)(ws);
  float*        mid   = (float*)(ws + (4u << 20));
  unsigned int* mb    = (unsigned int*)(ws + (8u << 20));
  float*        s1    = (float*)(ws + (16u << 20));
  float*        s2    = (float*)(ws + (16u << 20) + (64u << 10));
  float*        rowM  = (float*)(ws + (16u << 20) + (128u << 10));
  float*        rowIZ = (float*)(ws + (16u << 20) + (192u << 10));

  gat_pack_mask<<<(NTOT * MUSED + 255) / 256, 256, 0, stream>>>(
      A1, A2, A3, A12, A13, A23, A21, A31, A32, mb);
  gat_wh<<<NTOT, 128, 0, stream>>>(h, Ws, ap, WhI, s1, s2);
  gat_rowstats<<<(2 * NTOT) / 8, 256, 0, stream>>>(s1, s2, mb, rowM, rowIZ);
  gat_attn_wmma<<<(NTOT / 16) * 2, 256, 0, stream>>>(WhI, s1, s2, rowM, rowIZ, mb, mid);
  gat_logsoftmax<<<NTOT / 8, 256, 0, stream>>>(mid, out);
}